// SecureOptimizedBlockReLU_49624052137992
// MI455X (gfx1250) — compile-verified
//
#include <hip/hip_runtime.h>
#include <stdint.h>

// ---------------------------------------------------------------------------
// BlockReLU for MI455X (gfx1250).
// activation: (B=32, C=128, H=128, W=128) fp32, contiguous.
// Channel classes (compile-time constant from BLOCK_SIZES):
//   c in [0,32)   : 1x1 blocks (elementwise relu-by-multiply)
//   c in [32,80)  : 2x2 block-sum sign mask
//   c in [80,112) : 4x4 block-sum sign mask
//   c in [112,128): identity copy  -> done via Tensor Data Mover (TDM)
// Memory-bound: ~536 MB traffic, ~23 us floor at 23.3 TB/s. NT hints since
// the stream greatly exceeds the 192 MB L2.
// ---------------------------------------------------------------------------

typedef float        f32x4 __attribute__((ext_vector_type(4)));
typedef float        f32x2 __attribute__((ext_vector_type(2)));
typedef unsigned int u32x4 __attribute__((ext_vector_type(4)));
typedef int          i32x4 __attribute__((ext_vector_type(4)));
typedef int          i32x8 __attribute__((ext_vector_type(8)));

#define BATCH 32
#define CHANS 128
#define HDIM  128
#define WDIM  128
#define PLANE (HDIM * WDIM)   // 16384 floats = 64 KB

__device__ __forceinline__ float bmask(float s) { return s >= 0.0f ? 1.0f : 0.0f; }

// ---------------------------------------------------------------------------
// Identity channels (c in [112,128)): pure 64 KB-per-plane memcpy.
// Use the Tensor Data Mover: one 128x128 fp32 tile global->LDS, wait on
// TENSORcnt, then LDS->global. One wave per workgroup issues the DMA
// (EXEC is ignored by tensor ops); no VALU / VGPR traffic for the copy.
// Toolchain: clang-23 lane => 6-arg builtin form
//   (u32x4 g0, i32x8 g1, i32x4 g2, i32x4 g3, i32x8 extra, i32 cpol)
// Groups 2/3 and the extra group are unused for a <=2D tensor: zero-filled.
// (Placed first in the file so the disasm snippet shows the TDM sequence.)
// ---------------------------------------------------------------------------
#if defined(__HIP_DEVICE_COMPILE__) && \
    __has_builtin(__builtin_amdgcn_tensor_load_to_lds) && \
    __has_builtin(__builtin_amdgcn_tensor_store_from_lds)
#define USE_TDM 1
#endif

__device__ __forceinline__ void wait_tensorcnt0()
{
#if defined(__HIP_DEVICE_COMPILE__) && __has_builtin(__builtin_amdgcn_s_wait_tensorcnt)
  __builtin_amdgcn_s_wait_tensorcnt(0);
#elif defined(__HIP_DEVICE_COMPILE__)
  asm volatile("s_wait_tensorcnt 0x0" ::: "memory");
#endif
}

__global__ __launch_bounds__(32) void ident_copy_tdm(const float* __restrict__ act,
                                                     float* __restrict__ out)
{
  const int b = blockIdx.x >> 4;
  const int c = 112 + (blockIdx.x & 15);
  const long long base = (long long)(b * CHANS + c) * PLANE;

#if defined(USE_TDM)
  __shared__ float sh[PLANE];              // 64 KB staging tile
  const unsigned long long ga = (unsigned long long)(const void*)(act + base);
  const unsigned long long gs = (unsigned long long)(void*)(out + base);
  // Low 32 bits of a flat shared pointer == LDS byte offset (aperture rule).
  const unsigned lds = (unsigned)(unsigned long long)(void*)&sh[0];

  // D# group 0: count=1 | lds_addr | global_addr[56:0] | type=2 at [127:126]
  u32x4 g0l = { 1u, lds, (unsigned)(ga & 0xFFFFFFFFu),
                (unsigned)((ga >> 32) & 0x01FFFFFFu) | 0x80000000u };
  u32x4 g0s = { 1u, lds, (unsigned)(gs & 0xFFFFFFFFu),
                (unsigned)((gs >> 32) & 0x01FFFFFFu) | 0x80000000u };
  // D# group 1: 2D tensor 128x128 fp32 (data_size code 2 = 4B), tile = whole
  // plane 128x128, tensor_dim0_stride = 128, no multicast / iterate / pad.
  i32x8 g1 = { (int)0x00020000,        // wg_mask=0 | data_size=2 (4 bytes)
               (int)(128u << 16),      // bits[63:48]  = tensor_dim0.lo16 = 128
               (int)(128u << 16),      // bits[79:64]=dim0.hi=0 | bits[95:80]=tensor_dim1.lo16=128
               (int)(128u << 16),      // bits[111:96]=dim1.hi=0 | bits[127:112]=tile_dim0=128
               (int)128,               // bits[143:128]=tile_dim1=128 | tile_dim2=0
               (int)128,               // tensor_dim0_stride[31:0] = 128
               0, 0 };                 // stride0.hi=0, tensor_dim1_stride=0
  i32x4 gz4 = { 0, 0, 0, 0 };          // groups 2/3 unused (<=2D tensor)
  i32x8 gz8 = { 0, 0, 0, 0, 0, 0, 0, 0 };

  __builtin_amdgcn_tensor_load_to_lds(g0l, g1, gz4, gz4, gz8, 0);
  wait_tensorcnt0();                   // LDS tile resident
  __builtin_amdgcn_tensor_store_from_lds(g0s, g1, gz4, gz4, gz8, 0);
  wait_tensorcnt0();                   // copy complete before wave exit
#else
  // Fallback: straight NT vector copy with the single wave.
  const f32x4* __restrict__ s4 = (const f32x4*)(act + base);
  f32x4* __restrict__ d4 = (f32x4*)(out + base);
  const int t = threadIdx.x;
#pragma unroll 4
  for (int j = 0; j < PLANE / 4 / 32; ++j) {
    const int i = t + j * 32;
    __builtin_nontemporal_store(__builtin_nontemporal_load(s4 + i), d4 + i);
  }
#endif
}

// ---------------------------------------------------------------------------
// Main compute kernel: one workgroup per (b, c) plane, c < 112.
// Branch on channel class is workgroup-uniform.
// ---------------------------------------------------------------------------
__global__ __launch_bounds__(256) void block_relu_main(const float* __restrict__ act,
                                                       float* __restrict__ out)
{
  const int plane = blockIdx.x;            // 0 .. 4095
  const int c = plane & (CHANS - 1);
  if (c >= 112) return;                    // identity planes handled by TDM kernel
  const long long base = (long long)plane * PLANE;
  const float* __restrict__ src = act + base;
  float* __restrict__ dst = out + base;
  const int t = threadIdx.x;

  if (c < 32) {
    // 1x1: mask = (elem >= 0); out = elem * mask. 16 f32x4 per thread.
    const f32x4* __restrict__ s4 = (const f32x4*)src;
    f32x4* __restrict__ d4 = (f32x4*)dst;
#pragma unroll
    for (int j = 0; j < 16; ++j) {
      const int i = t + j * 256;
      f32x4 a = __builtin_nontemporal_load(s4 + i);
      f32x4 m = { bmask(a.x), bmask(a.y), bmask(a.z), bmask(a.w) };
      __builtin_nontemporal_store(a * m, d4 + i);
    }
  } else if (c < 80) {
    // 2x2: 64x64 = 4096 blocks; 16 blocks per thread; two f32x2 rows per block.
#pragma unroll
    for (int j = 0; j < 16; ++j) {
      const int bid = t + j * 256;
      const int by = bid >> 6, bx = bid & 63;
      const int o = (by * 2) * WDIM + bx * 2;
      f32x2 a0 = __builtin_nontemporal_load((const f32x2*)(src + o));
      f32x2 a1 = __builtin_nontemporal_load((const f32x2*)(src + o + WDIM));
      const float s = ((a0.x + a0.y) + a1.x) + a1.y;   // row-major block order
      const float m = bmask(s);
      __builtin_nontemporal_store(a0 * m, (f32x2*)(dst + o));
      __builtin_nontemporal_store(a1 * m, (f32x2*)(dst + o + WDIM));
    }
  } else {
    // 4x4: 32x32 = 1024 blocks; 4 blocks per thread; four f32x4 rows per block.
#pragma unroll
    for (int j = 0; j < 4; ++j) {
      const int bid = t + j * 256;
      const int by = bid >> 5, bx = bid & 31;
      const int o = (by * 4) * WDIM + bx * 4;
      f32x4 a0 = __builtin_nontemporal_load((const f32x4*)(src + o));
      f32x4 a1 = __builtin_nontemporal_load((const f32x4*)(src + o + WDIM));
      f32x4 a2 = __builtin_nontemporal_load((const f32x4*)(src + o + 2 * WDIM));
      f32x4 a3 = __builtin_nontemporal_load((const f32x4*)(src + o + 3 * WDIM));
      f32x4 ts = (a0 + a1) + (a2 + a3);
      const float s = (ts.x + ts.y) + (ts.z + ts.w);
      const float m = bmask(s);
      __builtin_nontemporal_store(a0 * m, (f32x4*)(dst + o));
      __builtin_nontemporal_store(a1 * m, (f32x4*)(dst + o + WDIM));
      __builtin_nontemporal_store(a2 * m, (f32x4*)(dst + o + 2 * WDIM));
      __builtin_nontemporal_store(a3 * m, (f32x4*)(dst + o + 3 * WDIM));
    }
  }
}

// ---------------------------------------------------------------------------
extern "C" void kernel_launch(void* const* d_in, const int* in_sizes, int n_in,
                              void* d_out, int out_size, void* d_ws, size_t ws_size,
                              hipStream_t stream)
{
  (void)in_sizes; (void)n_in; (void)d_ws; (void)ws_size; (void)out_size;
  const float* act = (const float*)d_in[0];
  float* out = (float*)d_out;

  // Compute classes: one workgroup per (b,c) plane; identity planes early-out.
  block_relu_main<<<dim3(BATCH * CHANS), dim3(256), 0, stream>>>(act, out);
  // Identity channels via TDM DMA: 32 batches x 16 channels, one wave each.
  ident_copy_tdm<<<dim3(BATCH * 16), dim3(32), 0, stream>>>(act, out);
}